// BackflowII_37692632989864
// MI455X (gfx1250) — compile-verified
//
#include <hip/hip_runtime.h>
#include <hip/hip_bf16.h>
#include <math.h>

// Problem constants (match reference)
#define BSZ  512
#define NSO  256
#define NF   128
#define HID  512
#define NCOL (NSO * NF)          // 32768 columns of F / W2
#define INV_NSITES 0.0078125f    // 128^-1

typedef __attribute__((ext_vector_type(2))) float v2f;
typedef __attribute__((ext_vector_type(8))) float v8f;

// ---------------------------------------------------------------------------
// Kernel 1: per-sample occupied-orbital scan + hidden layer
//   pos[b][s]  = exclusive prefix count of ones (row index in Asub if n[b][s]=1)
//   Rarr[b][i] = s-index of i-th occupied orbital (ascending, matches nonzero())
//   Hmat[b][h] = tanh(sum_{s in R} W1[s][h] + b1[h])
// ---------------------------------------------------------------------------
__global__ void setup_kernel(const int* __restrict__ n,
                             const float* __restrict__ W1,
                             const float* __restrict__ b1,
                             float* __restrict__ Hmat,
                             int* __restrict__ pos,
                             int* __restrict__ Rarr) {
    __shared__ int sscan[NSO];
    __shared__ int Rsh[NF];
    const int t = threadIdx.x;        // 0..255
    const int b = blockIdx.x;         // sample

    if (t < NF) Rsh[t] = 0;           // safety init (row always has NF ones)
    const int v = n[b * NSO + t];
    sscan[t] = v;
    for (int off = 1; off < NSO; off <<= 1) {
        __syncthreads();
        int add = (t >= off) ? sscan[t - off] : 0;
        __syncthreads();
        sscan[t] += add;
    }
    __syncthreads();
    const int ex = sscan[t] - v;      // exclusive prefix sum
    pos[b * NSO + t] = ex;
    if (v) {
        Rsh[ex] = t;
        Rarr[b * NF + ex] = t;
    }
    __syncthreads();

    // hidden layer: each thread handles h = t and t + 256 (coalesced over W1 rows)
    for (int h = t; h < HID; h += 256) {
        float acc = 0.0f;
        #pragma unroll 4
        for (int i = 0; i < NF; ++i)
            acc += W1[(size_t)Rsh[i] * HID + h];
        Hmat[(size_t)b * HID + h] = tanhf(acc + b1[h]);
    }
}

// ---------------------------------------------------------------------------
// Kernel 2: F = Hmat @ W2 (+ b2), fp32 WMMA 16x16x4, fused gather epilogue:
//   for selected orbital rows r of sample b:
//     Asub[b][pos[b][r]][f] = M[r][f] * (1 + F[b][r*NF+f] / 128)
// Grid: x = 256 column blocks (128 cols each), y = 32 sample groups (16 each).
// Workgroup: 256 threads = 8 waves; each wave owns one 16x16 C tile.
// ---------------------------------------------------------------------------
#define APAD 516   // 516 % 64 = 4 -> conflict-free lane-strided A reads
__global__ void fgemm_kernel(const float* __restrict__ Hmat,
                             const float* __restrict__ W2,
                             const float* __restrict__ b2,
                             const float* __restrict__ Mmat,
                             const int* __restrict__ n,
                             const int* __restrict__ pos,
                             float* __restrict__ Asub) {
    __shared__ float Ash[16][APAD];           // 16 x 512 A tile (padded)
    const int t    = threadIdx.x;             // 0..255
    const int wave = t >> 5;                  // 0..7
    const int lane = t & 31;
    const int bg   = blockIdx.y;              // sample group

    // stage A tile (16 samples x K=512) into LDS
    for (int idx = t; idx < 16 * HID; idx += 256) {
        int m = idx >> 9, k = idx & (HID - 1);
        Ash[m][k] = Hmat[((size_t)bg * 16 + m) * HID + k];
    }
    __syncthreads();

    // WMMA fragment lane mapping (ISA 32-bit A 16x4 / B 4x16 layouts)
    const int l15  = lane & 15;               // M row (A) / N col (B)
    const int kap  = (lane >> 4) * 2;         // K sub-offset: 0 or 2
    const int ncol = blockIdx.x * 128 + wave * 16 + l15;  // global F column

    v8f acc = {};
    for (int k0 = 0; k0 < HID; k0 += 4) {
        v2f a, bfrag;
        a.x     = Ash[l15][k0 + kap];
        a.y     = Ash[l15][k0 + kap + 1];
        bfrag.x = W2[(size_t)(k0 + kap)     * NCOL + ncol];
        bfrag.y = W2[(size_t)(k0 + kap + 1) * NCOL + ncol];
        acc = __builtin_amdgcn_wmma_f32_16x16x4_f32(
            false, a, false, bfrag, (short)0, acc, false, false);
    }

    // epilogue: this wave's 16 columns all belong to one orbital r
    const int c0    = blockIdx.x * 128 + wave * 16;
    const int r_orb = c0 >> 7;                // column / NF  (wave-uniform)
    const int f     = (c0 & (NF - 1)) + l15;  // feature index
    const float mv  = Mmat[(size_t)r_orb * NF + f];
    const float b2v = b2[ncol];
    #pragma unroll
    for (int r = 0; r < 8; ++r) {
        int m  = r + (lane >> 4) * 8;         // C layout: lanes16-31 hold M = r+8
        int bs = bg * 16 + m;
        if (n[bs * NSO + r_orb]) {
            int i = pos[bs * NSO + r_orb];
            float Fv = acc[r] + b2v;
            Asub[((size_t)bs * NF + i) * NF + f] = mv * (1.0f + Fv * INV_NSITES);
        }
    }
}

// ---------------------------------------------------------------------------
// Kernel 3: slogdet via partial-pivoted Gaussian elimination in LDS.
// One 128-thread workgroup per sample; matrix in dynamic LDS (128 x 129 f32).
// out[b] = complex64: (log|det|, sign < 0 ? pi : 0)
// ---------------------------------------------------------------------------
#define LUP 129
__global__ void lu_kernel(const float* __restrict__ Asub,
                          float* __restrict__ out) {
    extern __shared__ float As[];             // 128*129 floats = 66048 B
    __shared__ float red_v[NF];
    __shared__ int   red_i[NF];
    __shared__ int   s_p;
    __shared__ float s_logabs;
    __shared__ int   s_sign;
    const int t = threadIdx.x;                // 0..127
    const int b = blockIdx.x;

    for (int c = 0; c < NF; ++c)
        As[t * LUP + c] = Asub[((size_t)b * NF + t) * NF + c];
    if (t == 0) { s_logabs = 0.0f; s_sign = 1; }
    __syncthreads();

    for (int k = 0; k < NF; ++k) {
        // partial pivot: argmax_j>=k |As[j][k]|
        red_v[t] = (t >= k) ? fabsf(As[t * LUP + k]) : -1.0f;
        red_i[t] = t;
        __syncthreads();
        for (int off = NF / 2; off > 0; off >>= 1) {
            if (t < off && red_v[t + off] > red_v[t]) {
                red_v[t] = red_v[t + off];
                red_i[t] = red_i[t + off];
            }
            __syncthreads();
        }
        if (t == 0) s_p = red_i[0];
        __syncthreads();
        const int p = s_p;
        if (p != k) {                          // swap rows k<->p, thread t = column
            float tmp = As[k * LUP + t];
            As[k * LUP + t] = As[p * LUP + t];
            As[p * LUP + t] = tmp;
        }
        __syncthreads();
        const float pv = As[k * LUP + k];
        if (t == 0) {
            if (p != k)    s_sign = -s_sign;   // row-swap parity
            if (pv < 0.0f) s_sign = -s_sign;   // pivot sign
            s_logabs += logf(fabsf(pv));
        }
        if (t > k) {                           // eliminate row t
            float fct = As[t * LUP + k] / pv;
            for (int c = k + 1; c < NF; ++c)
                As[t * LUP + c] -= fct * As[k * LUP + c];
        }
        __syncthreads();
    }
    if (t == 0) {
        out[b * 2 + 0] = s_logabs;
        out[b * 2 + 1] = (s_sign < 0) ? 3.14159265358979323846f : 0.0f;
    }
}

// ---------------------------------------------------------------------------
extern "C" void kernel_launch(void* const* d_in, const int* in_sizes, int n_in,
                              void* d_out, int out_size, void* d_ws, size_t ws_size,
                              hipStream_t stream) {
    const int*   n   = (const int*)  d_in[0];
    const float* W1  = (const float*)d_in[1];
    const float* b1  = (const float*)d_in[2];
    const float* W2  = (const float*)d_in[3];
    const float* b2  = (const float*)d_in[4];
    const float* Mm  = (const float*)d_in[5];

    char* ws = (char*)d_ws;
    float* Hmat = (float*)(ws);                                   // 512*512*4   = 1 MB
    int*   pos  = (int*)  (ws + (1u << 20));                      // 512*256*4   = 512 KB
    int*   Rarr = (int*)  (ws + (1u << 20) + (512u << 10));       // 512*128*4   = 256 KB
    float* Asub = (float*)(ws + (2u << 20));                      // 512*128*128*4 = 32 MB

    setup_kernel<<<BSZ, 256, 0, stream>>>(n, W1, b1, Hmat, pos, Rarr);

    dim3 g2(NCOL / 128, BSZ / 16);            // 256 x 32 workgroups
    fgemm_kernel<<<g2, 256, 0, stream>>>(Hmat, W2, b2, Mm, n, pos, Asub);

    lu_kernel<<<BSZ, NF, NF * LUP * sizeof(float), stream>>>(Asub, (float*)d_out);
}